// QuanvolutionClassifier_65481071403242
// MI455X (gfx1250) — compile-verified
//
#include <hip/hip_runtime.h>
#include <hip/hip_bf16.h>
#include <math.h>

// ---------------------------------------------------------------------------
// Kernel 1: quantum circuit simulation. One batch element per lane; the 16
// complex amplitudes live in 32 VGPRs. All gate loops are fully unrolled with
// compile-time bit masks (templates) so no array ever hits scratch.
// ---------------------------------------------------------------------------

// wire w <-> bit (3-w) of the flat 4-bit state index (wire 0 = MSB)

template <int MASK>
__device__ __forceinline__ void apply_ry(float* ar, float* ai, float c, float s) {
#pragma unroll
  for (int k = 0; k < 16; ++k) {
    if (!(k & MASK)) {
      const int k1 = k | MASK;
      float a0r = ar[k], a0i = ai[k], a1r = ar[k1], a1i = ai[k1];
      ar[k]  = c * a0r - s * a1r;  ai[k]  = c * a0i - s * a1i;
      ar[k1] = s * a0r + c * a1r;  ai[k1] = s * a0i + c * a1i;
    }
  }
}

template <int MASK>
__device__ __forceinline__ void apply_rz(float* ar, float* ai, float c, float s) {
  // bit==0 amp *= (c - i s); bit==1 amp *= (c + i s)
#pragma unroll
  for (int k = 0; k < 16; ++k) {
    const float sg = (k & MASK) ? s : -s;
    float r = ar[k], im = ai[k];
    ar[k] = c * r - sg * im;
    ai[k] = c * im + sg * r;
  }
}

template <int MASK>
__device__ __forceinline__ void apply_rx(float* ar, float* ai, float c, float s) {
#pragma unroll
  for (int k = 0; k < 16; ++k) {
    if (!(k & MASK)) {
      const int k1 = k | MASK;
      float a0r = ar[k], a0i = ai[k], a1r = ar[k1], a1i = ai[k1];
      // new0 = c*a0 - i s*a1 ; new1 = -i s*a0 + c*a1
      ar[k]  =  c * a0r + s * a1i;  ai[k]  =  c * a0i - s * a1r;
      ar[k1] =  s * a0i + c * a1r;  ai[k1] = -s * a0r + c * a1i;
    }
  }
}

template <int MC, int MT>
__device__ __forceinline__ void apply_cnot(float* ar, float* ai) {
#pragma unroll
  for (int k = 0; k < 16; ++k) {
    if ((k & MC) && !(k & MT)) {
      const int k1 = k | MT;
      float t;
      t = ar[k]; ar[k] = ar[k1]; ar[k1] = t;
      t = ai[k]; ai[k] = ai[k1]; ai[k1] = t;
    }
  }
}

__global__ __launch_bounds__(256) void quanv_kernel(
    const float* __restrict__ x,        // [B, 28, 28]
    const float* __restrict__ w_enc,    // [4, 4]
    const float* __restrict__ rz,       // [2, 4]
    const float* __restrict__ rx,       // [2, 4]
    float* __restrict__ feats)          // [B, 784]
{
  const int b = blockIdx.x * 256 + threadIdx.x;

  // fixed-gate trig (uniform across lanes -> scalar loads + cheap VALU)
  float cz[8], sz[8], cxv[8], sxv[8];
#pragma unroll
  for (int i = 0; i < 8; ++i) {
    float hz = 0.5f * rz[i], hx = 0.5f * rx[i];
    cz[i] = cosf(hz); sz[i] = sinf(hz);
    cxv[i] = cosf(hx); sxv[i] = sinf(hx);
  }
  float we[16];
#pragma unroll
  for (int i = 0; i < 16; ++i) we[i] = w_enc[i];

  float ar[16], ai[16];
#pragma unroll
  for (int k = 0; k < 16; ++k) { ar[k] = 0.0f; ai[k] = 0.0f; }
  ar[0] = 1.0f;

  const float* img = x + (size_t)b * 784;
  float* fb = feats + (size_t)b * 784;

  int pidx = 0;
#pragma unroll 1
  for (int pi = 0; pi < 14; ++pi) {
    const float* r0 = img + pi * 56;   // row 2*pi
#pragma unroll 1
    for (int pj = 0; pj < 14; ++pj) {
      const float v0 = r0[2 * pj], v1 = r0[2 * pj + 1];
      const float v2 = r0[2 * pj + 28], v3 = r0[2 * pj + 29];

      // angles[j] = sum_i patch[i] * w_enc[j][i]
      float ang0 = we[0] * v0 + we[1] * v1 + we[2] * v2 + we[3] * v3;
      float ang1 = we[4] * v0 + we[5] * v1 + we[6] * v2 + we[7] * v3;
      float ang2 = we[8] * v0 + we[9] * v1 + we[10] * v2 + we[11] * v3;
      float ang3 = we[12] * v0 + we[13] * v1 + we[14] * v2 + we[15] * v3;

      // data-dependent RY on each wire
      apply_ry<8>(ar, ai, cosf(0.5f * ang0), sinf(0.5f * ang0));
      apply_ry<4>(ar, ai, cosf(0.5f * ang1), sinf(0.5f * ang1));
      apply_ry<2>(ar, ai, cosf(0.5f * ang2), sinf(0.5f * ang2));
      apply_ry<1>(ar, ai, cosf(0.5f * ang3), sinf(0.5f * ang3));

      // 2 fixed layers: per-wire RZ,RX then ring of CNOTs
#pragma unroll
      for (int l = 0; l < 2; ++l) {
        const int o = l * 4;
        apply_rz<8>(ar, ai, cz[o + 0], sz[o + 0]); apply_rx<8>(ar, ai, cxv[o + 0], sxv[o + 0]);
        apply_rz<4>(ar, ai, cz[o + 1], sz[o + 1]); apply_rx<4>(ar, ai, cxv[o + 1], sxv[o + 1]);
        apply_rz<2>(ar, ai, cz[o + 2], sz[o + 2]); apply_rx<2>(ar, ai, cxv[o + 2], sxv[o + 2]);
        apply_rz<1>(ar, ai, cz[o + 3], sz[o + 3]); apply_rx<1>(ar, ai, cxv[o + 3], sxv[o + 3]);
        apply_cnot<8, 4>(ar, ai);   // (0,1)
        apply_cnot<4, 2>(ar, ai);   // (1,2)
        apply_cnot<2, 1>(ar, ai);   // (2,3)
        apply_cnot<1, 8>(ar, ai);   // (3,0)
      }

      // <Z_w> measurement
      float m0 = 0.f, m1 = 0.f, m2 = 0.f, m3 = 0.f;
#pragma unroll
      for (int k = 0; k < 16; ++k) {
        const float pk = ar[k] * ar[k] + ai[k] * ai[k];
        m0 += (k & 8) ? -pk : pk;
        m1 += (k & 4) ? -pk : pk;
        m2 += (k & 2) ? -pk : pk;
        m3 += (k & 1) ? -pk : pk;
      }
      float4 mv = make_float4(m0, m1, m2, m3);
      *reinterpret_cast<float4*>(fb + pidx * 4) = mv;
      ++pidx;
    }
  }
}

// ---------------------------------------------------------------------------
// Kernel 2: classifier GEMM with V_WMMA_F32_16X16X4_F32 + fused log_softmax.
// One wave owns one 16(M)x16(N) f32 tile; K=784 in 196 WMMA steps.
// ---------------------------------------------------------------------------

typedef __attribute__((ext_vector_type(2))) float v2f;
typedef __attribute__((ext_vector_type(8))) float v8f;

__global__ __launch_bounds__(256) void cls_kernel(
    const float* __restrict__ feats,   // [B, 784]
    const float* __restrict__ w_cls,   // [10, 784]
    const float* __restrict__ b_cls,   // [10]
    float* __restrict__ out,           // [B, 10]
    int B)
{
  __shared__ float Bpad[16 * 784];     // w_cls zero-padded to 16 rows (B^T panel)
  __shared__ float Ctile[8][16][17];   // per-wave C tiles (padded stride)

  const int tid = threadIdx.x;

  // stage padded w_cls panel into LDS
  for (int idx = tid; idx < 16 * 784; idx += 256) {
    const int n = idx / 784;
    const int k = idx - n * 784;
    Bpad[idx] = (n < 10) ? w_cls[n * 784 + k] : 0.0f;
  }
  __syncthreads();

  const int lane = tid & 31;
  const int wave = tid >> 5;
  const int half = lane >> 4;       // K-pair select: lanes 0-15 -> K{0,1}, 16-31 -> K{2,3}
  const int lr   = lane & 15;       // M row (A) / N col (B)
  const int m0   = (blockIdx.x * 8 + wave) * 16;

  const float* arow = feats + (size_t)(m0 + lr) * 784 + 2 * half;
  const float* brow = Bpad + lr * 784 + 2 * half;

  v8f acc0 = {};
  v8f acc1 = {};
#pragma unroll 4
  for (int k = 0; k < 784; k += 8) {
    v2f a0 = *reinterpret_cast<const v2f*>(arow + k);
    v2f b0 = *reinterpret_cast<const v2f*>(brow + k);
    acc0 = __builtin_amdgcn_wmma_f32_16x16x4_f32(false, a0, false, b0,
                                                 (short)0, acc0, false, false);
    v2f a1 = *reinterpret_cast<const v2f*>(arow + k + 4);
    v2f b1 = *reinterpret_cast<const v2f*>(brow + k + 4);
    acc1 = __builtin_amdgcn_wmma_f32_16x16x4_f32(false, a1, false, b1,
                                                 (short)0, acc1, false, false);
  }

  // C layout: VGPR j, lanes 0-15 -> M=j, N=lane; lanes 16-31 -> M=8+j, N=lane-16
#pragma unroll
  for (int j = 0; j < 8; ++j)
    Ctile[wave][j + 8 * half][lr] = acc0[j] + acc1[j];
  __syncthreads();

  // 128 rows per block; threads 0..127 each finish one row (bias + log_softmax)
  if (tid < 128) {
    const int wv = tid >> 4;
    const int mr = tid & 15;
    const int row = blockIdx.x * 128 + tid;   // == (blockIdx.x*8 + wv)*16 + mr
    float lg[10];
    float mx = -3.4e38f;
#pragma unroll
    for (int n = 0; n < 10; ++n) {
      lg[n] = Ctile[wv][mr][n] + b_cls[n];
      mx = fmaxf(mx, lg[n]);
    }
    float se = 0.0f;
#pragma unroll
    for (int n = 0; n < 10; ++n) se += expf(lg[n] - mx);
    const float lse = mx + logf(se);
    if (row < B) {
#pragma unroll
      for (int n = 0; n < 10; ++n) out[row * 10 + n] = lg[n] - lse;
    }
  }
}

// ---------------------------------------------------------------------------

extern "C" void kernel_launch(void* const* d_in, const int* in_sizes, int n_in,
                              void* d_out, int out_size, void* d_ws, size_t ws_size,
                              hipStream_t stream) {
  const float* x     = (const float*)d_in[0];   // [B,28,28]
  const float* w_enc = (const float*)d_in[1];   // [4,4]
  const float* rz    = (const float*)d_in[2];   // [2,4]
  const float* rx    = (const float*)d_in[3];   // [2,4]
  const float* w_cls = (const float*)d_in[4];   // [10,784]
  const float* b_cls = (const float*)d_in[5];   // [10]
  float* out = (float*)d_out;
  float* feats = (float*)d_ws;                  // [B,784] intermediate (51 MB, lives in L2)

  const int B = in_sizes[0] / 784;              // 16384

  quanv_kernel<<<B / 256, 256, 0, stream>>>(x, w_enc, rz, rx, feats);
  cls_kernel<<<B / 128, 256, 0, stream>>>(feats, w_cls, b_cls, out, B);
}